// GATRoutingModel_31009663877407
// MI455X (gfx1250) — compile-verified
//
#include <hip/hip_runtime.h>

#define NEG_SLOPE 0.2f

typedef float v2f __attribute__((ext_vector_type(2)));
typedef float v8f __attribute__((ext_vector_type(8)));

__device__ __forceinline__ float leaky(float v) {
    return v > 0.0f ? v : NEG_SLOPE * v;
}

// float atomic-max via integer ordering trick (valid for mixed-sign floats)
__device__ __forceinline__ void atomicMaxF(float* addr, float v) {
    if (v >= 0.0f) atomicMax((int*)addr, __float_as_int(v));
    else           atomicMin((unsigned int*)addr, __float_as_uint(v));
}

// ---------------------------------------------------------------------------
// Out[N,C] = X[N,K] @ W[K,C] using V_WMMA_F32_16X16X4_F32.
// One wave (32 lanes) owns one 16x16 output tile; blockDim = (32,8) = 8 waves.
// K and C are compile-time so every load/store in the fully unrolled loop is
// a single base pointer + immediate offset (no per-iteration 64-bit muls).
// A layout (16x4 f32): lanes 0-15 -> K = k0..k0+1, lanes 16-31 -> K = k0+2..k0+3
// B layout (4x16 f32): VGPR0 = row k0 (+2 for upper half-lanes), VGPR1 = row k0+1
// C/D layout: VGPR v -> M = v + 8*(lane>=16), N = lane&15
// ---------------------------------------------------------------------------
template<int K, int C>
__global__ void gemm_wmma(const float* __restrict__ X, const float* __restrict__ W,
                          float* __restrict__ Out, int N) {
    const int tileM = blockIdx.x * blockDim.y + threadIdx.y;
    if (tileM * 16 >= N) return;                 // wave-uniform: EXEC stays all-ones
    const int tileN = blockIdx.y;
    const int lane  = threadIdx.x;
    const int half  = lane >> 4;
    const int l16   = lane & 15;
    const int row   = tileM * 16 + l16;
    const int col   = tileN * 16 + l16;

    const float* __restrict__ ap = X + row * K + half * 2;       // + k0        per step
    const float* __restrict__ bp = W + (half * 2) * C + col;     // + k0*C      per step
    float*       __restrict__ op = Out + (tileM * 16 + half * 8) * C + col;

    v8f acc = {};
#pragma unroll
    for (int k0 = 0; k0 < K; k0 += 4) {
        v2f a = *(const v2f*)(ap + k0);          // global_load_b64, imm offset
        v2f b;
        b.x = bp[k0 * C];                        // global_load_b32, imm offset
        b.y = bp[k0 * C + C];
        acc = __builtin_amdgcn_wmma_f32_16x16x4_f32(
            /*neg_a=*/false, a, /*neg_b=*/false, b,
            /*c_mod=*/(short)0, acc, /*reuse_a=*/false, /*reuse_b=*/false);
    }
#pragma unroll
    for (int v = 0; v < 8; ++v)
        op[v * C] = acc[v];                      // global_store_b32, imm offset
}

// ---------------------------------------------------------------------------
// Per-node: alpha_src/alpha_dst dots; init running max m with the self-loop
// logit (self loops always present -> no -inf init pass needed).
// ---------------------------------------------------------------------------
__global__ void node_alpha_init(const float* __restrict__ Htmp,
                                const float* __restrict__ a_src,
                                const float* __restrict__ a_dst,
                                float* __restrict__ as, float* __restrict__ ad,
                                float* __restrict__ m, int N, int H, int D) {
    int i = blockIdx.x * blockDim.x + threadIdx.x;
    if (i >= N) return;
    for (int h = 0; h < H; ++h) {
        const float* hp = Htmp + ((size_t)i * H + h) * D;
        float ss = 0.f, sd = 0.f;
        for (int d = 0; d < D; ++d) {
            float v = hp[d];
            ss += v * a_src[h * D + d];
            sd += v * a_dst[h * D + d];
        }
        as[(size_t)i * H + h] = ss;
        ad[(size_t)i * H + h] = sd;
        m[(size_t)i * H + h]  = leaky(ss + sd);   // self-loop logit seeds the max
    }
}

// Pass 1 over edges: running max of leaky(logit) per destination/head.
__global__ void edge_max(const int* __restrict__ src, const int* __restrict__ dst,
                         const float* __restrict__ as, const float* __restrict__ ad,
                         float* __restrict__ m, int E, int H) {
    int e = blockIdx.x * blockDim.x + threadIdx.x;
    if (e >= E) return;
    int s = src[e], d = dst[e];
    for (int h = 0; h < H; ++h) {
        float v = leaky(as[(size_t)s * H + h] + ad[(size_t)d * H + h]);
        atomicMaxF(&m[(size_t)d * H + h], v);
    }
}

// Per-node: self-loop exp term initializes denom and the (unnormalized) agg.
__global__ void node_self(const float* __restrict__ Htmp,
                          const float* __restrict__ as, const float* __restrict__ ad,
                          const float* __restrict__ m,
                          float* __restrict__ denom, float* __restrict__ agg,
                          int N, int H, int D) {
    int i = blockIdx.x * blockDim.x + threadIdx.x;
    if (i >= N) return;
    for (int h = 0; h < H; ++h) {
        float e = leaky(as[(size_t)i * H + h] + ad[(size_t)i * H + h]);
        float p = expf(e - m[(size_t)i * H + h]);
        denom[(size_t)i * H + h] = p;
        const float* hp = Htmp + ((size_t)i * H + h) * D;
        float*       ag = agg  + ((size_t)i * H + h) * D;
        for (int d = 0; d < D; ++d) ag[d] = p * hp[d];
    }
}

// Pass 2 over edges: p = exp(logit - max); denom += p; agg[dst] += p * h[src].
// p is recomputed, never stored. Gathers hit L2 (feature table ~12.8MB << 192MB).
__global__ void edge_agg(const int* __restrict__ src, const int* __restrict__ dst,
                         const float* __restrict__ Htmp,
                         const float* __restrict__ as, const float* __restrict__ ad,
                         const float* __restrict__ m,
                         float* __restrict__ denom, float* __restrict__ agg,
                         int E, int H, int D) {
    int e = blockIdx.x * blockDim.x + threadIdx.x;
    if (e >= E) return;
    int s = src[e], d = dst[e];
    for (int h = 0; h < H; ++h) {
        float v = leaky(as[(size_t)s * H + h] + ad[(size_t)d * H + h]);
        float p = expf(v - m[(size_t)d * H + h]);
        atomicAdd(&denom[(size_t)d * H + h], p);
        const float4* hs = (const float4*)(Htmp + ((size_t)s * H + h) * D);
        float*        ag = agg + ((size_t)d * H + h) * D;
        for (int q = 0; q < D / 4; ++q) {
            float4 x = hs[q];                      // global_load_b128 gather
            atomicAdd(ag + 4 * q + 0, p * x.x);
            atomicAdd(ag + 4 * q + 1, p * x.y);
            atomicAdd(ag + 4 * q + 2, p * x.z);
            atomicAdd(ag + 4 * q + 3, p * x.w);
        }
    }
}

// out = elu(agg/denom + bias)
__global__ void finalize(const float* __restrict__ agg, const float* __restrict__ denom,
                         const float* __restrict__ bias, float* __restrict__ out,
                         int N, int H, int D) {
    int idx = blockIdx.x * blockDim.x + threadIdx.x;
    if (idx >= N * H * D) return;
    int i = idx / (H * D);
    int j = idx - i * (H * D);
    int h = j / D;
    float v = agg[idx] / denom[(size_t)i * H + h] + bias[j];
    out[idx] = v > 0.f ? v : (expf(v) - 1.f);
}

// scores[i] = h2[i,:] . fcW + fcb
__global__ void fc_scores(const float* __restrict__ h2, const float* __restrict__ fcW,
                          const float* __restrict__ fcb, float* __restrict__ scores,
                          int N, int D) {
    int i = blockIdx.x * blockDim.x + threadIdx.x;
    if (i >= N) return;
    float s = fcb[0];
    for (int d = 0; d < D; ++d) s += h2[(size_t)i * D + d] * fcW[d];
    scores[i] = s;
}

extern "C" void kernel_launch(void* const* d_in, const int* in_sizes, int n_in,
                              void* d_out, int out_size, void* d_ws, size_t ws_size,
                              hipStream_t stream) {
    const float* x      = (const float*)d_in[0];
    const int*   ei     = (const int*)  d_in[1];
    const float* W1     = (const float*)d_in[2];
    const float* a_src1 = (const float*)d_in[3];
    const float* a_dst1 = (const float*)d_in[4];
    const float* b1     = (const float*)d_in[5];
    const float* W2     = (const float*)d_in[6];
    const float* a_src2 = (const float*)d_in[7];
    const float* a_dst2 = (const float*)d_in[8];
    const float* b2     = (const float*)d_in[9];
    const float* fcW    = (const float*)d_in[10];
    const float* fcb    = (const float*)d_in[11];

    const int N = in_sizes[0] / 64;     // F_in = 64
    const int E = in_sizes[1] / 2;      // edge_index is [2, E]
    const int* src = ei;
    const int* dst = ei + E;

    // workspace carve-out (floats); total = N*268 floats ~54 MB
    float* p     = (float*)d_ws;
    float* h1tmp = p; p += (size_t)N * 64;
    float* as1   = p; p += (size_t)N * 2;
    float* ad1   = p; p += (size_t)N * 2;
    float* m1    = p; p += (size_t)N * 2;
    float* den1  = p; p += (size_t)N * 2;
    float* agg1  = p; p += (size_t)N * 64;
    float* h1    = p; p += (size_t)N * 64;
    float* h2tmp = p; p += (size_t)N * 32;
    float* as2   = p; p += (size_t)N;
    float* ad2   = p; p += (size_t)N;
    float* m2    = p; p += (size_t)N;
    float* den2  = p; p += (size_t)N;
    float* agg2  = p; p += (size_t)N * 32;

    float* scores = (float*)d_out;       // [N]
    float* h2out  = scores + N;          // [N,32]

    const int tilesM = (N + 15) / 16;
    const dim3 gblk(32, 8);              // 8 waves per block, 1 tile per wave
    const int WPB = 8;
    const int TB = 256;
    const int nodeGrid = (N + TB - 1) / TB;
    const int edgeGrid = (E + TB - 1) / TB;

    // ---------------- Layer 1: GATConv(64 -> 2 heads x 32, concat) ----------
    gemm_wmma<64, 64><<<dim3((tilesM + WPB - 1) / WPB, 64 / 16), gblk, 0, stream>>>(x, W1, h1tmp, N);
    node_alpha_init<<<nodeGrid, TB, 0, stream>>>(h1tmp, a_src1, a_dst1, as1, ad1, m1, N, 2, 32);
    edge_max<<<edgeGrid, TB, 0, stream>>>(src, dst, as1, ad1, m1, E, 2);
    node_self<<<nodeGrid, TB, 0, stream>>>(h1tmp, as1, ad1, m1, den1, agg1, N, 2, 32);
    edge_agg<<<edgeGrid, TB, 0, stream>>>(src, dst, h1tmp, as1, ad1, m1, den1, agg1, E, 2, 32);
    finalize<<<(N * 64 + TB - 1) / TB, TB, 0, stream>>>(agg1, den1, b1, h1, N, 2, 32);

    // ---------------- Layer 2: GATConv(64 -> 1 head x 32, mean) -------------
    gemm_wmma<64, 32><<<dim3((tilesM + WPB - 1) / WPB, 32 / 16), gblk, 0, stream>>>(h1, W2, h2tmp, N);
    node_alpha_init<<<nodeGrid, TB, 0, stream>>>(h2tmp, a_src2, a_dst2, as2, ad2, m2, N, 1, 32);
    edge_max<<<edgeGrid, TB, 0, stream>>>(src, dst, as2, ad2, m2, E, 1);
    node_self<<<nodeGrid, TB, 0, stream>>>(h2tmp, as2, ad2, m2, den2, agg2, N, 1, 32);
    edge_agg<<<edgeGrid, TB, 0, stream>>>(src, dst, h2tmp, as2, ad2, m2, den2, agg2, E, 1, 32);
    finalize<<<(N * 32 + TB - 1) / TB, TB, 0, stream>>>(agg2, den2, b2, h2out, N, 1, 32);

    // ---------------- FC head ------------------------------------------------
    fc_scores<<<nodeGrid, TB, 0, stream>>>(h2out, fcW, fcb, scores, N, 32);
}